// SampleGenerateGT_69999376990263
// MI455X (gfx1250) — compile-verified
//
#include <hip/hip_runtime.h>

// Bilinear gather: out[s, c] = bilerp(data, xs[s])  for 16 channels.
// Memory-bound (AI ~0.5 flop/byte): optimize purely for the gather path.
//   - 4 lanes per sample, one float4 channel-group per lane -> all b128 VMEM.
//   - NT hints on streaming xs reads / out writes to preserve L2 for the texture.

typedef float v4f __attribute__((ext_vector_type(4)));
typedef float v2f __attribute__((ext_vector_type(2)));

#define IMG_H 2048
#define IMG_W 2048
#define NSAMP 4194304u

__global__ __launch_bounds__(256)
void bilerp_gather_kernel(const v2f* __restrict__ xs2,   // N float2 coords
                          const v4f* __restrict__ data4, // H*W*4 float4 (16 ch/pixel)
                          v4f* __restrict__ out4)        // N*4 float4
{
    const unsigned gid = blockIdx.x * 256u + threadIdx.x;
    const unsigned s   = gid >> 2;    // sample index (shared by 4 lanes)
    const unsigned cg  = gid & 3u;    // channel group: floats [4*cg, 4*cg+4)

    // Streaming read of coords (NT: read-once data, don't pollute L2).
    // The 4 lanes of a group hit the same cacheline; L0 merges them.
    v2f xy = __builtin_nontemporal_load(xs2 + s);

    float sx = xy.x * (float)IMG_W;
    float sy = xy.y * (float)IMG_H;
    int ix = (int)sx;
    int iy = (int)sy;
    float wx = sx - (float)ix;
    float wy = sy - (float)iy;

    int x0 = min(max(ix, 0), IMG_W - 1);
    int y0 = min(max(iy, 0), IMG_H - 1);
    int x1 = min(x0 + 1, IMG_W - 1);
    int y1 = min(y0 + 1, IMG_H - 1);

    // float4-granular offsets; max index < 2048*2048*4 = 2^24 -> 32-bit math,
    // lets the compiler use SGPR-base + 32-bit VGPR-offset global addressing.
    unsigned r0 = (unsigned)y0 * (IMG_W * 4u);
    unsigned r1 = (unsigned)y1 * (IMG_W * 4u);
    unsigned c0 = (unsigned)x0 * 4u + cg;
    unsigned c1 = (unsigned)x1 * 4u + cg;

    // Four b128 gathers; issued back-to-back so LOADcnt overlaps all four.
    // Texture uses default RT caching: ~4x reuse and footprint ~ L2 size.
    v4f v00 = data4[r0 + c0];
    v4f v01 = data4[r0 + c1];
    v4f v10 = data4[r1 + c0];
    v4f v11 = data4[r1 + c1];

    // Same weight formulation as the reference.
    float w00 = (1.0f - wx) * (1.0f - wy);
    float w01 = wx * (1.0f - wy);
    float w10 = (1.0f - wx) * wy;
    float w11 = wx * wy;

    v4f acc = v00 * w00 + v01 * w01 + v10 * w10 + v11 * w11;

    // Streaming b128 store (NT): output is write-once, keep it out of L2.
    __builtin_nontemporal_store(acc, out4 + (s * 4u + cg));
}

extern "C" void kernel_launch(void* const* d_in, const int* in_sizes, int n_in,
                              void* d_out, int out_size, void* d_ws, size_t ws_size,
                              hipStream_t stream) {
    const v2f* xs2   = (const v2f*)d_in[0];   // xs: (N, 2) f32
    const v4f* data4 = (const v4f*)d_in[1];   // data: (H, W, 16) f32
    v4f* out4        = (v4f*)d_out;           // out: (N, 16) f32

    const unsigned total_threads = NSAMP * 4u;        // 16,777,216
    dim3 grid(total_threads / 256u), block(256u);
    bilerp_gather_kernel<<<grid, block, 0, stream>>>(xs2, data4, out4);
}